// PCFLayer_20255065768445
// MI455X (gfx1250) — compile-verified
//
#include <hip/hip_runtime.h>
#include <math.h>

// ---------------------------------------------------------------------------
// PointConvFormer layer for MI455X (gfx1250).
// fp32 WMMA (V_WMMA_F32_16X16X4_F32) + GLOBAL_LOAD_ASYNC_TO_LDS_B128 gathers.
// ---------------------------------------------------------------------------

typedef float v2f __attribute__((ext_vector_type(2)));
typedef float v8f __attribute__((ext_vector_type(8)));

#define WMMA4(a, b, c) \
  __builtin_amdgcn_wmma_f32_16x16x4_f32(false, (a), false, (b), (short)0, (c), false, false)

__device__ __forceinline__ float leaky01(float v) { return v > 0.f ? v : 0.1f * v; }
__device__ __forceinline__ float relu1(float v)   { return v > 0.f ? v : 0.f; }
__device__ __forceinline__ float sigm(float v)    { return 1.f / (1.f + __expf(-v)); }

// 64-byte async copy global -> LDS (4 x b128), ASYNCcnt-tracked.
// INST_OFFSET applies to both LDS and global sides (ISA 10.x async pseudocode).
__device__ __forceinline__ void async_copy64B(unsigned ldsa, const float* src) {
  asm volatile(
      "global_load_async_to_lds_b128 %0, %1, off\n\t"
      "global_load_async_to_lds_b128 %0, %1, off offset:16\n\t"
      "global_load_async_to_lds_b128 %0, %1, off offset:32\n\t"
      "global_load_async_to_lds_b128 %0, %1, off offset:48"
      :: "v"(ldsa), "v"(src) : "memory");
}
__device__ __forceinline__ void wait_async0() { asm volatile("s_wait_asynccnt 0x0" ::: "memory"); }
__device__ __forceinline__ void wait_ds0()    { asm volatile("s_wait_dscnt 0x0" ::: "memory"); }

constexpr int N_PTS = 200000;
constexpr int M_PTS = 100000;

// ===========================================================================
// Kernel A: feats_x = leaky(dense_feats @ w1 + b1); guid = feats_x @ wg + bg
// ===========================================================================
__global__ __launch_bounds__(256) void unary_guidance_kernel(
    const float* __restrict__ df,   // [N,64]
    const float* __restrict__ w1,   // [64,32]
    const float* __restrict__ b1,   // [32]
    const float* __restrict__ wg,   // [32,32]
    const float* __restrict__ bg,   // [32]
    float* __restrict__ feats_x,    // [N,32]
    float* __restrict__ guid,       // [N,32]
    int n)
{
  __shared__ v2f  sW1p[32 * 32];     // row-pair-packed [64,32]
  __shared__ v2f  sWGp[16 * 32];     // row-pair-packed [32,32]
  __shared__ float sB1[32];
  __shared__ float sBG[32];
  __shared__ float sA[8][16 * 34];   // per-wave staging, stride 34 keeps pairs 8B-aligned

  const int tid = threadIdx.x;
  for (int i = tid; i < 32 * 32; i += 256) {
    int r2 = i >> 5, c = i & 31;
    v2f v; v.x = w1[(2 * r2) * 32 + c]; v.y = w1[(2 * r2 + 1) * 32 + c];
    sW1p[i] = v;
  }
  for (int i = tid; i < 16 * 32; i += 256) {
    int r2 = i >> 5, c = i & 31;
    v2f v; v.x = wg[(2 * r2) * 32 + c]; v.y = wg[(2 * r2 + 1) * 32 + c];
    sWGp[i] = v;
  }
  if (tid < 32) { sB1[tid] = b1[tid]; sBG[tid] = bg[tid]; }
  __syncthreads();

  const int wave = tid >> 5, lane = tid & 31;
  const int lo = lane & 15, hi = lane >> 4;
  const int base = (blockIdx.x * 8 + wave) * 16;
  if (base >= n) return;                       // wave-uniform guard

  // ---- unary1: [16,64] @ [64,32] ----
  v8f acc0 = {}, acc1 = {};
  const float* arow = df + (size_t)(base + lo) * 64;
  #pragma unroll
  for (int k0 = 0; k0 < 64; k0 += 4) {
    const int kk = k0 + 2 * hi;
    const v2f a  = *(const v2f*)(arow + kk);
    const v2f b0 = sW1p[(kk >> 1) * 32 + lo];
    const v2f b1v = sW1p[(kk >> 1) * 32 + 16 + lo];
    acc0 = WMMA4(a, b0, acc0);
    acc1 = WMMA4(a, b1v, acc1);
  }
  float* sa = sA[wave];
  #pragma unroll
  for (int r = 0; r < 8; ++r) {
    const int row = r + 8 * hi;
    const float v0 = leaky01(acc0[r] + sB1[lo]);
    const float v1 = leaky01(acc1[r] + sB1[16 + lo]);
    feats_x[(size_t)(base + row) * 32 + lo]      = v0;
    feats_x[(size_t)(base + row) * 32 + 16 + lo] = v1;
    sa[row * 34 + lo]      = v0;
    sa[row * 34 + 16 + lo] = v1;
  }
  __builtin_amdgcn_wave_barrier();

  // ---- guidance_unary: [16,32] @ [32,32] ----
  v8f g0 = {}, g1 = {};
  #pragma unroll
  for (int k0 = 0; k0 < 32; k0 += 4) {
    const int kk = k0 + 2 * hi;
    const v2f a  = *(const v2f*)&sa[lo * 34 + kk];
    const v2f b0 = sWGp[(kk >> 1) * 32 + lo];
    const v2f b1v = sWGp[(kk >> 1) * 32 + 16 + lo];
    g0 = WMMA4(a, b0, g0);
    g1 = WMMA4(a, b1v, g1);
  }
  #pragma unroll
  for (int r = 0; r < 8; ++r) {
    const int row = r + 8 * hi;
    guid[(size_t)(base + row) * 32 + lo]      = g0[r] + sBG[lo];
    guid[(size_t)(base + row) * 32 + 16 + lo] = g1[r] + sBG[16 + lo];
  }
}

// ===========================================================================
// Kernel P: pair-pack the big back-end weights into workspace
//   wlinp[(k/2)*64 + c]  = {wlin[k][c], wlin[k+1][c]}      (512x64)
//   wu2p / wscp likewise for the 64x128 weights
// ===========================================================================
__global__ __launch_bounds__(256) void pack_pairs_kernel(
    const float* __restrict__ wlin, const float* __restrict__ wu2,
    const float* __restrict__ wsc,
    v2f* __restrict__ wlinp, v2f* __restrict__ wu2p, v2f* __restrict__ wscp)
{
  const int i = blockIdx.x * 256 + threadIdx.x;
  if (i < 256 * 64) {
    int r2 = i >> 6, c = i & 63;
    v2f v; v.x = wlin[(2 * r2) * 64 + c]; v.y = wlin[(2 * r2 + 1) * 64 + c];
    wlinp[i] = v;
  }
  if (i < 32 * 128) {
    int r2 = i >> 7, c = i & 127;
    v2f v; v.x = wu2[(2 * r2) * 128 + c]; v.y = wu2[(2 * r2 + 1) * 128 + c];
    wu2p[i] = v;
    v2f s; s.x = wsc[(2 * r2) * 128 + c]; s.y = wsc[(2 * r2 + 1) * 128 + c];
    wscp[i] = s;
  }
}

// ===========================================================================
// Kernel B: neighborhood stage + back-end GEMMs.
//   64 threads = 2 waves; 16 sparse points per block.
// ===========================================================================
__global__ __launch_bounds__(64) void pcf_main_kernel(
    const float* __restrict__ df,     // dense_feats [N,64]
    const float* __restrict__ vif,    // vi_features [M,16,12]
    const int*   __restrict__ nei,    // nei_inds [M,16]
    const float* __restrict__ feats,  // feats_x [N,32]
    const float* __restrict__ guid,   // guidance_x [N,32]
    const float* __restrict__ wpe, const float* __restrict__ bpe,
    const float* __restrict__ wg1, const float* __restrict__ bg1,
    const float* __restrict__ wg2, const float* __restrict__ bg2,
    const float* __restrict__ wn1, const float* __restrict__ bn1,
    const float* __restrict__ wn2, const float* __restrict__ bn2,
    const float* __restrict__ wn3, const float* __restrict__ bn3,
    const v2f* __restrict__ wlinp, const float* __restrict__ blin,
    const v2f* __restrict__ wu2p,  const float* __restrict__ bu2,
    const v2f* __restrict__ wscp,  const float* __restrict__ bsc,
    float* __restrict__ out)          // [M,128]
{
  // ---- staged small weights, row-pair-packed, zero-padded to 16-col tiles ----
  __shared__ v2f sWPEp[6 * 32];
  __shared__ float sBPE[32];
  __shared__ v2f sWG1p[32 * 16];
  __shared__ float sBG1[16];
  __shared__ v2f sWG2p[8 * 16];
  __shared__ float sBG2[16];
  __shared__ v2f sWN1p[6 * 16];
  __shared__ float sBN1[16];
  __shared__ v2f sWN2p[4 * 16];
  __shared__ float sBN2[16];
  __shared__ v2f sWN3p[4 * 16];
  __shared__ float sBN3[16];
  // ---- per-wave front-end scratch ----
  __shared__ float sGF[2][16 * 64];    // g_feat [k][64] (cols 0..31 async-gathered)
  __shared__ float sNF[2][16 * 32];    // raw gathered feats_x (async); aliased as sOUT later
  __shared__ float sT1[2][16 * 16];    // weightnet intermediate / pair-packed wgt
  __shared__ float sT2[2][16 * 16];    // guidance MLP intermediate / scores
  __shared__ float sCM[2][64];         // column max of g_feat
  __shared__ int   sNID[2][16];
  // ---- per-block tile buffers ----
  __shared__ float sAGG[16][512];      // einsum result per point
  __shared__ float sSF[16][64];        // shortcut max-pooled feats

  const int tid = threadIdx.x;
  const int wave = tid >> 5, lane = tid & 31;
  const int lo = lane & 15, hi = lane >> 4;
  const int m0 = blockIdx.x * 16;

  if (tid == 0) {
    __builtin_prefetch(wlinp, 0, 1);
    __builtin_prefetch(wu2p, 0, 1);
    __builtin_prefetch(wscp, 0, 1);
  }

  for (int i = tid; i < 6 * 32; i += 64) {
    int r2 = i >> 5, c = i & 31;
    v2f v; v.x = wpe[(2 * r2) * 32 + c]; v.y = wpe[(2 * r2 + 1) * 32 + c];
    sWPEp[i] = v;
  }
  for (int i = tid; i < 32 * 16; i += 64) {
    int r2 = i >> 4, c = i & 15;
    v2f v = {};
    if (c < 8) { v.x = wg1[(2 * r2) * 8 + c]; v.y = wg1[(2 * r2 + 1) * 8 + c]; }
    sWG1p[i] = v;
  }
  for (int i = tid; i < 8 * 16; i += 64) {
    int r2 = i >> 4, c = i & 15;
    v2f v = {};
    if (r2 < 4 && c < 8) { v.x = wg2[(2 * r2) * 8 + c]; v.y = wg2[(2 * r2 + 1) * 8 + c]; }
    sWG2p[i] = v;
  }
  for (int i = tid; i < 6 * 16; i += 64) {
    int r2 = i >> 4, c = i & 15;
    v2f v = {};
    if (c < 8) { v.x = wn1[(2 * r2) * 8 + c]; v.y = wn1[(2 * r2 + 1) * 8 + c]; }
    sWN1p[i] = v;
  }
  for (int i = tid; i < 4 * 16; i += 64) {
    int r2 = i >> 4, c = i & 15;
    v2f v = {};
    if (c < 8) { v.x = wn2[(2 * r2) * 8 + c]; v.y = wn2[(2 * r2 + 1) * 8 + c]; }
    sWN2p[i] = v;
  }
  for (int i = tid; i < 4 * 16; i += 64) {
    int r2 = i >> 4, c = i & 15;
    v2f v; v.x = wn3[(2 * r2) * 16 + c]; v.y = wn3[(2 * r2 + 1) * 16 + c];
    sWN3p[i] = v;
  }
  if (tid < 32) sBPE[tid] = bpe[tid];
  if (tid < 16) {
    sBG1[tid] = tid < 8 ? bg1[tid] : 0.f;
    sBG2[tid] = tid < 8 ? bg2[tid] : 0.f;
    sBN1[tid] = tid < 8 ? bn1[tid] : 0.f;
    sBN2[tid] = tid < 8 ? bn2[tid] : 0.f;
    sBN3[tid] = bn3[tid];
  }
  __syncthreads();

  // =========================== front-end =================================
  float* gf  = sGF[wave];
  float* nf  = sNF[wave];
  float* t1  = sT1[wave];
  float* t2  = sT2[wave];
  float* cm  = sCM[wave];
  int*   nid = sNID[wave];
  v2f*   t1p = (v2f*)t1;

  for (int p = wave * 8; p < wave * 8 + 8; ++p) {
    const int m = m0 + p;

    // make sure the previous iteration's LDS reads retired before the async
    // engine can overwrite gf/nf (async writes are unordered vs DS ops)
    wait_ds0();

    if (lane < 16) nid[lane] = nei[(size_t)m * 16 + lane];
    __builtin_amdgcn_wave_barrier();

    // async gathers (ASYNCcnt): guidance rows -> gf[:,0:32], feats_x -> nf
    {
      const int k = lane >> 1, cb = (lane & 1) * 16;
      async_copy64B((unsigned)(uintptr_t)(gf + k * 64 + cb),
                    guid + (size_t)nid[k] * 32 + cb);
      async_copy64B((unsigned)(uintptr_t)(nf + k * 32 + cb),
                    feats + (size_t)nid[k] * 32 + cb);
    }

    const float* vi = vif + (size_t)m * 16 * 12;   // [16,12]

    // feat_pe = relu(vi @ wpe + bpe) -> gf[k][32..63]   (overlaps async)
    {
      v8f pe0 = {}, pe1 = {};
      #pragma unroll
      for (int k0 = 0; k0 < 12; k0 += 4) {
        const int kk = k0 + 2 * hi;
        const v2f a  = *(const v2f*)(vi + lo * 12 + kk);
        const v2f b0 = sWPEp[(kk >> 1) * 32 + lo];
        const v2f b1 = sWPEp[(kk >> 1) * 32 + 16 + lo];
        pe0 = WMMA4(a, b0, pe0);
        pe1 = WMMA4(a, b1, pe1);
      }
      #pragma unroll
      for (int r = 0; r < 8; ++r) {
        const int row = r + 8 * hi;
        gf[row * 64 + 32 + lo] = relu1(pe0[r] + sBPE[lo]);
        gf[row * 64 + 48 + lo] = relu1(pe1[r] + sBPE[16 + lo]);
      }
    }

    // WeightNet: vi -> 8 -> 8 -> 16 (relu); final wgt pair-packed into t1p
    {
      v8f w = {};
      #pragma unroll
      for (int k0 = 0; k0 < 12; k0 += 4) {
        const int kk = k0 + 2 * hi;
        const v2f a = *(const v2f*)(vi + lo * 12 + kk);
        const v2f b = sWN1p[(kk >> 1) * 16 + lo];
        w = WMMA4(a, b, w);
      }
      #pragma unroll
      for (int r = 0; r < 8; ++r) t1[(r + 8 * hi) * 16 + lo] = relu1(w[r] + sBN1[lo]);
      __builtin_amdgcn_wave_barrier();

      v8f w2 = {};
      #pragma unroll
      for (int k0 = 0; k0 < 8; k0 += 4) {
        const int kk = k0 + 2 * hi;
        const v2f a = *(const v2f*)&t1[lo * 16 + kk];
        const v2f b = sWN2p[(kk >> 1) * 16 + lo];
        w2 = WMMA4(a, b, w2);
      }
      #pragma unroll
      for (int r = 0; r < 8; ++r) t1[(r + 8 * hi) * 16 + lo] = relu1(w2[r] + sBN2[lo]);
      __builtin_amdgcn_wave_barrier();

      v8f w3 = {};
      #pragma unroll
      for (int k0 = 0; k0 < 8; k0 += 4) {
        const int kk = k0 + 2 * hi;
        const v2f a = *(const v2f*)&t1[lo * 16 + kk];
        const v2f b = sWN3p[(kk >> 1) * 16 + lo];
        w3 = WMMA4(a, b, w3);
      }
      // consecutive accumulator rows r, r+1 live in this lane: pack pairs
      #pragma unroll
      for (int r = 0; r < 8; r += 2) {
        v2f v; v.x = relu1(w3[r] + sBN3[lo]); v.y = relu1(w3[r + 1] + sBN3[lo]);
        t1p[((r + 8 * hi) >> 1) * 16 + lo] = v;
      }
    }

    wait_async0();               // gf[:,0:32] and nf are now valid
    __builtin_amdgcn_wave_barrier();

    // column max over the 16 neighbors (2 cols per lane)
    {
      const int c = lane * 2;
      float mx0 = -3.0e38f, mx1 = -3.0e38f;
      #pragma unroll
      for (int k = 0; k < 16; ++k) {
        mx0 = fmaxf(mx0, gf[k * 64 + c]);
        mx1 = fmaxf(mx1, gf[k * 64 + c + 1]);
      }
      cm[c] = mx0; cm[c + 1] = mx1;
    }
    __builtin_amdgcn_wave_barrier();

    // guidance MLP: relu((g_feat - colmax) @ wg1 + b) -> sigmoid(@ wg2 + b)
    {
      v8f s1 = {};
      #pragma unroll
      for (int k0 = 0; k0 < 64; k0 += 4) {
        const int kk = k0 + 2 * hi;
        const v2f g2 = *(const v2f*)&gf[lo * 64 + kk];
        const v2f c2 = *(const v2f*)&cm[kk];
        const v2f a = g2 - c2;
        const v2f b = sWG1p[(kk >> 1) * 16 + lo];
        s1 = WMMA4(a, b, s1);
      }
      #pragma unroll
      for (int r = 0; r < 8; ++r) t2[(r + 8 * hi) * 16 + lo] = relu1(s1[r] + sBG1[lo]);
      __builtin_amdgcn_wave_barrier();

      v8f s2 = {};
      #pragma unroll
      for (int k0 = 0; k0 < 8; k0 += 4) {
        const int kk = k0 + 2 * hi;
        const v2f a = *(const v2f*)&t2[lo * 16 + kk];
        const v2f b = sWG2p[(kk >> 1) * 16 + lo];
        s2 = WMMA4(a, b, s2);
      }
      #pragma unroll
      for (int r = 0; r < 8; ++r) t2[(r + 8 * hi) * 16 + lo] = sigm(s2[r] + sBG2[lo]);
    }
    __builtin_amdgcn_wave_barrier();

    // einsum: agg[c][w] = sum_k (nf[k][c] * score[k][c/4]) * wgt[k][w]
    #pragma unroll
    for (int ct = 0; ct < 2; ++ct) {
      const int c = ct * 16 + lo;
      v8f ag = {};
      #pragma unroll
      for (int k0 = 0; k0 < 16; k0 += 4) {
        const int kk = k0 + 2 * hi;
        v2f a;
        a.x = nf[kk * 32 + c]       * t2[kk * 16 + (c >> 2)];
        a.y = nf[(kk + 1) * 32 + c] * t2[(kk + 1) * 16 + (c >> 2)];
        const v2f b = t1p[(kk >> 1) * 16 + lo];
        ag = WMMA4(a, b, ag);
      }
      #pragma unroll
      for (int r = 0; r < 8; ++r) {
        const int cc = ct * 16 + r + 8 * hi;
        sAGG[p][cc * 16 + lo] = ag[r];
      }
    }

    // shortcut: max over neighborhood of dense_feats (2 cols per lane)
    {
      float mx0 = -3.0e38f, mx1 = -3.0e38f;
      #pragma unroll
      for (int k = 0; k < 16; ++k) {
        const float* dfr = df + (size_t)nid[k] * 64;
        mx0 = fmaxf(mx0, dfr[lane]);
        mx1 = fmaxf(mx1, dfr[lane + 32]);
      }
      sSF[p][lane] = mx0;
      sSF[p][lane + 32] = mx1;
    }
  }
  __syncthreads();

  // =========================== back-end =================================
  float* sOUT = &sNF[0][0];   // 16*64 floats, front-end scratch is dead now

  // out = relu(agg[16,512] @ wlin[512,64] + blin)  : 2 n-tiles per wave
  for (int t = 0; t < 2; ++t) {
    const int nt = wave * 2 + t;
    v8f acc = {};
    for (int k0 = 0; k0 < 512; k0 += 4) {
      const int kk = k0 + 2 * hi;
      const v2f a = *(const v2f*)&sAGG[lo][kk];
      const v2f b = wlinp[(size_t)(kk >> 1) * 64 + nt * 16 + lo];
      acc = WMMA4(a, b, acc);
    }
    #pragma unroll
    for (int r = 0; r < 8; ++r) {
      const int row = r + 8 * hi, col = nt * 16 + lo;
      sOUT[row * 64 + col] = relu1(acc[r] + blin[col]);
    }
  }
  __syncthreads();

  // final = leaky(out @ wu2 + bu2 + sf @ wsc + bsc) : 4 n-tiles per wave
  for (int t = 0; t < 4; ++t) {
    const int nt = wave * 4 + t;
    v8f acc = {};
    for (int k0 = 0; k0 < 64; k0 += 4) {
      const int kk = k0 + 2 * hi;
      const v2f a  = *(const v2f*)&sOUT[lo * 64 + kk];
      const v2f b  = wu2p[(size_t)(kk >> 1) * 128 + nt * 16 + lo];
      acc = WMMA4(a, b, acc);
      const v2f a2 = *(const v2f*)&sSF[lo][kk];
      const v2f b2 = wscp[(size_t)(kk >> 1) * 128 + nt * 16 + lo];
      acc = WMMA4(a2, b2, acc);
    }
    #pragma unroll
    for (int r = 0; r < 8; ++r) {
      const int row = r + 8 * hi, col = nt * 16 + lo;
      out[(size_t)(m0 + row) * 128 + col] = leaky01(acc[r] + bu2[col] + bsc[col]);
    }
  }
}

// ===========================================================================
extern "C" void kernel_launch(void* const* d_in, const int* in_sizes, int n_in,
                              void* d_out, int out_size, void* d_ws, size_t ws_size,
                              hipStream_t stream) {
  (void)in_sizes; (void)n_in; (void)out_size; (void)ws_size;

  const float* dense_feats = (const float*)d_in[1];
  const float* vi          = (const float*)d_in[5];
  const int*   nei         = (const int*)d_in[6];
  const float* w_unary1 = (const float*)d_in[7];
  const float* b_unary1 = (const float*)d_in[8];
  const float* w_gu     = (const float*)d_in[9];
  const float* b_gu     = (const float*)d_in[10];
  const float* w_pe     = (const float*)d_in[11];
  const float* b_pe     = (const float*)d_in[12];
  const float* w_g1     = (const float*)d_in[13];
  const float* b_g1     = (const float*)d_in[14];
  const float* w_g2     = (const float*)d_in[15];
  const float* b_g2     = (const float*)d_in[16];
  const float* w_wn1    = (const float*)d_in[17];
  const float* b_wn1    = (const float*)d_in[18];
  const float* w_wn2    = (const float*)d_in[19];
  const float* b_wn2    = (const float*)d_in[20];
  const float* w_wn3    = (const float*)d_in[21];
  const float* b_wn3    = (const float*)d_in[22];
  const float* w_lin    = (const float*)d_in[23];
  const float* b_lin    = (const float*)d_in[24];
  const float* w_u2     = (const float*)d_in[25];
  const float* b_u2     = (const float*)d_in[26];
  const float* w_sc     = (const float*)d_in[27];
  const float* b_sc     = (const float*)d_in[28];

  float* feats_x = (float*)d_ws;                       // [N,32]
  float* guid    = feats_x + (size_t)N_PTS * 32;       // [N,32]
  v2f*   wlinp   = (v2f*)(guid + (size_t)N_PTS * 32);  // 256*64 pairs
  v2f*   wu2p    = wlinp + 256 * 64;                   // 32*128 pairs
  v2f*   wscp    = wu2p + 32 * 128;                    // 32*128 pairs
  float* outp    = (float*)d_out;                      // [M,128]

  pack_pairs_kernel<<<64, 256, 0, stream>>>(w_lin, w_u2, w_sc, wlinp, wu2p, wscp);

  const int tilesA = N_PTS / 16;                       // 12500
  unary_guidance_kernel<<<(tilesA + 7) / 8, 256, 0, stream>>>(
      dense_feats, w_unary1, b_unary1, w_gu, b_gu, feats_x, guid, N_PTS);

  pcf_main_kernel<<<M_PTS / 16, 64, 0, stream>>>(
      dense_feats, vi, nei, feats_x, guid,
      w_pe, b_pe, w_g1, b_g1, w_g2, b_g2,
      w_wn1, b_wn1, w_wn2, b_wn2, w_wn3, b_wn3,
      wlinp, b_lin, wu2p, b_u2, wscp, b_sc, outp);
}